// AttentionDecoderRNN_77077483094124
// MI455X (gfx1250) — compile-verified
//
#include <hip/hip_runtime.h>
#include <math.h>

#define B_ 256
#define S_ 256
#define H_ 512
#define E_ 256
#define V_ 32000

typedef float v2f __attribute__((ext_vector_type(2)));
typedef float v8f __attribute__((ext_vector_type(8)));

// D = A(16x4 f32) * B(4x16 f32) + C   -- V_WMMA_F32_16X16X4_F32 (wave32)
__device__ __forceinline__ v8f wmma4(v2f a, v2f b, v8f c) {
  return __builtin_amdgcn_wmma_f32_16x16x4_f32(false, a, false, b, (short)0, c,
                                               false, false);
}

// 32x64 output macro-tile of C = A * W^T per wave (2 m-frags x 4 n-frags).
// Per K-step: 6 fragment loads feed 8 WMMAs. Register footprint ~110 VGPRs
// (8 x v8f accumulators + frags + pointers) -- fits without scratch spills.
// Fragment layout (ISA 7.12.2, 32-bit A 16x4 / B 4x16):
//   lane&15 selects M (A) / N (B); lane>>4 selects K pair {0,1} vs {2,3}.
__device__ __forceinline__ void wmma_gemm_32x64(const float* __restrict__ A, int lda,
                                                const float* __restrict__ W, int ldw,
                                                int K, v8f (&acc)[2][4]) {
  const int lane = threadIdx.x & 31;
  const int idx  = lane & 15;
  const int half = lane >> 4;
  const float* ap[2];
  const float* wp[4];
#pragma unroll
  for (int i = 0; i < 2; i++)
    ap[i] = A + (i * 16 + idx) * lda + half * 2;
#pragma unroll
  for (int j = 0; j < 4; j++) {
    wp[j] = W + (j * 16 + idx) * ldw + half * 2;
    __builtin_prefetch(wp[j], 0, 1);  // global_prefetch_b8: warm weight stream
  }
  for (int k = 0; k < K; k += 4) {
    v2f a[2], b[4];
#pragma unroll
    for (int i = 0; i < 2; i++) { a[i].x = ap[i][k]; a[i].y = ap[i][k + 1]; }
#pragma unroll
    for (int j = 0; j < 4; j++) { b[j].x = wp[j][k]; b[j].y = wp[j][k + 1]; }
#pragma unroll
    for (int i = 0; i < 2; i++)
#pragma unroll
      for (int j = 0; j < 4; j++)
        acc[i][j] = wmma4(a[i], b[j], acc[i][j]);
  }
}

// ---------------------------------------------------------------------------
// K1: gather embedded token vectors and h0 into the fused x buffer
// ws_x layout per row b (1280 floats): [context(512) | embedded(256) | h0(512)]
// ---------------------------------------------------------------------------
__global__ void k_embed_prep(const int* __restrict__ tokens,
                             const float* __restrict__ hidden,
                             const float* __restrict__ emb,
                             float* __restrict__ ws_x) {
  const int b = blockIdx.x;
  const int t = threadIdx.x;  // 256
  float* x = ws_x + b * 1280;
  x[512 + t]       = emb[tokens[b] * E_ + t];
  x[768 + t]       = hidden[b * H_ + t];
  x[768 + 256 + t] = hidden[b * H_ + 256 + t];
}

// ---------------------------------------------------------------------------
// K2: u[b,h] = h0 @ W_energy[:, :H]^T + b_energy      (M=256, N=512, K=512)
// grid (2,8), block 128: each wave owns a 32x64 macro-tile
// ---------------------------------------------------------------------------
__global__ void __launch_bounds__(128, 1)
k_gemm_u(const float* __restrict__ hidden,
         const float* __restrict__ W_energy,
         const float* __restrict__ b_energy,
         float* __restrict__ ws_u) {
  const int wave  = threadIdx.x >> 5;
  const int nbase = (blockIdx.x * 4 + wave) * 64;
  const int mbase = blockIdx.y * 32;
  v8f acc[2][4] = {};
  wmma_gemm_32x64(hidden + mbase * H_, H_,
                  W_energy + nbase * (2 * H_), 2 * H_, H_, acc);
  const int lane = threadIdx.x & 31, idx = lane & 15, half = lane >> 4;
#pragma unroll
  for (int j = 0; j < 4; j++) {
    const int col = nbase + j * 16 + idx;
    const float bias = b_energy[col];
#pragma unroll
    for (int i = 0; i < 2; i++)
#pragma unroll
      for (int r = 0; r < 8; r++) {
        const int row = mbase + i * 16 + r + 8 * half;
        ws_u[row * H_ + col] = acc[i][j][r] + bias;
      }
  }
}

// ---------------------------------------------------------------------------
// K3: score[b,s] = sum_h w_value[h] * tanh( enc[b,s]·We2[h] + u[b,h] )
// Fused: the [B,S,H] energy tensor is never written to memory.
// grid 4096 (one block = 16 rows of flattened (b,s)), block 256.
// Each wave owns 4 N-tiles (64 h columns); A fragment reused across them.
// ---------------------------------------------------------------------------
__global__ void k_score(const float* __restrict__ enc,
                        const float* __restrict__ W_energy,
                        const float* __restrict__ ws_u,
                        const float* __restrict__ w_value,
                        float* __restrict__ ws_score) {
  const int rowbase = blockIdx.x * 16;     // flattened (b*S + s) tile start
  const int b       = rowbase >> 8;        // S_ == 256, 16 | 256 -> single b
  const int wave = threadIdx.x >> 5;
  const int lane = threadIdx.x & 31;
  const int idx  = lane & 15;
  const int half = lane >> 4;

  const float* ap = enc + (size_t)rowbase * H_ + idx * H_ + half * 2;
  const float* bp[4];
#pragma unroll
  for (int nt = 0; nt < 4; nt++) {
    bp[nt] = W_energy + ((wave * 4 + nt) * 16 + idx) * (2 * H_) + H_ + half * 2;
    __builtin_prefetch(bp[nt], 0, 1);
  }

  v8f acc[4] = {};
  for (int k = 0; k < H_; k += 4) {
    v2f a; a.x = ap[k]; a.y = ap[k + 1];
#pragma unroll
    for (int nt = 0; nt < 4; nt++) {
      v2f bb; bb.x = bp[nt][k]; bb.y = bp[nt][k + 1];
      acc[nt] = wmma4(a, bb, acc[nt]);
    }
  }

  __shared__ float s_score[16];
  if (threadIdx.x < 16) s_score[threadIdx.x] = 0.f;
  __syncthreads();

  float rs[8];
#pragma unroll
  for (int r = 0; r < 8; r++) rs[r] = 0.f;

#pragma unroll
  for (int nt = 0; nt < 4; nt++) {
    const int col = (wave * 4 + nt) * 16 + idx;
    const float uu  = ws_u[b * H_ + col];
    const float wvv = w_value[col];
#pragma unroll
    for (int r = 0; r < 8; r++) {
      float v = tanhf(acc[nt][r] + uu) * wvv;
      // butterfly sum over the 16 columns held by this half-wave
      v += __shfl_xor(v, 8, 16);
      v += __shfl_xor(v, 4, 16);
      v += __shfl_xor(v, 2, 16);
      v += __shfl_xor(v, 1, 16);
      rs[r] += v;
    }
  }
  if (idx == 0) {
#pragma unroll
    for (int r = 0; r < 8; r++) atomicAdd(&s_score[r + 8 * half], rs[r]);
  }
  __syncthreads();
  if (threadIdx.x < 16) ws_score[rowbase + threadIdx.x] = s_score[threadIdx.x];
}

// ---------------------------------------------------------------------------
// K4: softmax over S per batch, write attn output, compute context into ws_x
// grid B, block 256 (one thread per s; then 2 h per thread for context)
// ---------------------------------------------------------------------------
__global__ void k_softmax_ctx(const float* __restrict__ ws_score,
                              const float* __restrict__ enc,
                              float* __restrict__ out_attn,
                              float* __restrict__ ws_x) {
  const int b = blockIdx.x;
  const int t = threadIdx.x;  // 256
  __shared__ float red[256];
  __shared__ float sa[256];

  const float v = ws_score[b * S_ + t];
  red[t] = v; __syncthreads();
  for (int off = 128; off; off >>= 1) {
    if (t < off) red[t] = fmaxf(red[t], red[t + off]);
    __syncthreads();
  }
  const float m = red[0]; __syncthreads();
  const float e = __expf(v - m);
  red[t] = e; __syncthreads();
  for (int off = 128; off; off >>= 1) {
    if (t < off) red[t] += red[t + off];
    __syncthreads();
  }
  const float a = e / red[0];
  sa[t] = a;
  out_attn[b * S_ + t] = a;
  __syncthreads();

  float c0 = 0.f, c1 = 0.f;
  const float* eb = enc + (size_t)b * S_ * H_;
  for (int s = 0; s < S_; s++) {
    const float av = sa[s];
    c0 += av * eb[s * H_ + t];
    c1 += av * eb[s * H_ + 256 + t];
  }
  ws_x[b * 1280 + t]       = c0;
  ws_x[b * 1280 + 256 + t] = c1;
}

// ---------------------------------------------------------------------------
// K5: gates = x @ W_ih^T + h0 @ W_hh^T + b_ih + b_hh   (M=256, N=2048)
// grid (8,8), block 128: 32x64 macro-tile per wave, two K segments
// ---------------------------------------------------------------------------
__global__ void __launch_bounds__(128, 1)
k_gemm_gates(const float* __restrict__ ws_x,
             const float* __restrict__ W_ih,
             const float* __restrict__ W_hh,
             const float* __restrict__ b_ih,
             const float* __restrict__ b_hh,
             float* __restrict__ ws_gates) {
  const int wave  = threadIdx.x >> 5;
  const int nbase = (blockIdx.x * 4 + wave) * 64;
  const int mbase = blockIdx.y * 32;
  v8f acc[2][4] = {};
  wmma_gemm_32x64(ws_x + mbase * 1280, 1280,
                  W_ih + nbase * 768, 768, 768, acc);            // x part (K=768)
  wmma_gemm_32x64(ws_x + mbase * 1280 + 768, 1280,
                  W_hh + nbase * H_, H_, H_, acc);               // h0 part (K=512)
  const int lane = threadIdx.x & 31, idx = lane & 15, half = lane >> 4;
#pragma unroll
  for (int j = 0; j < 4; j++) {
    const int col = nbase + j * 16 + idx;
    const float bias = b_ih[col] + b_hh[col];
#pragma unroll
    for (int i = 0; i < 2; i++)
#pragma unroll
      for (int r = 0; r < 8; r++) {
        const int row = mbase + i * 16 + r + 8 * half;
        ws_gates[row * 2048 + col] = acc[i][j][r] + bias;
      }
  }
}

// ---------------------------------------------------------------------------
// K6: LSTM pointwise (PyTorch gate order i,f,g,o)
// ---------------------------------------------------------------------------
__device__ __forceinline__ float sigm(float x) { return 1.f / (1.f + __expf(-x)); }

__global__ void k_lstm(const float* __restrict__ ws_gates,
                       const float* __restrict__ cell,
                       float* __restrict__ out_h,
                       float* __restrict__ out_c) {
  const int gid = blockIdx.x * 256 + threadIdx.x;  // [0, B*H)
  const int b = gid >> 9;
  const int h = gid & 511;
  const float* g = ws_gates + b * 2048;
  const float ig = sigm(g[h]);
  const float fg = sigm(g[512 + h]);
  const float gg = tanhf(g[1024 + h]);
  const float og = sigm(g[1536 + h]);
  const float c  = fg * cell[b * H_ + h] + ig * gg;
  out_c[b * H_ + h] = c;
  out_h[b * H_ + h] = og * tanhf(c);
}

// ---------------------------------------------------------------------------
// K7: logits = relu(h_new @ W_out^T + b_out)           (M=256, N=32000, K=512)
// grid (125,8), block 128: 32x64 macro-tile per wave
// ---------------------------------------------------------------------------
__global__ void __launch_bounds__(128, 1)
k_gemm_logits(const float* __restrict__ h_new,
              const float* __restrict__ W_out,
              const float* __restrict__ b_out,
              float* __restrict__ out_logits) {
  const int wave  = threadIdx.x >> 5;
  const int nbase = (blockIdx.x * 4 + wave) * 64;
  const int mbase = blockIdx.y * 32;
  v8f acc[2][4] = {};
  wmma_gemm_32x64(h_new + mbase * H_, H_,
                  W_out + nbase * H_, H_, H_, acc);
  const int lane = threadIdx.x & 31, idx = lane & 15, half = lane >> 4;
#pragma unroll
  for (int j = 0; j < 4; j++) {
    const int col = nbase + j * 16 + idx;
    const float bias = b_out[col];
#pragma unroll
    for (int i = 0; i < 2; i++)
#pragma unroll
      for (int r = 0; r < 8; r++) {
        const int row = mbase + i * 16 + r + 8 * half;
        out_logits[row * V_ + col] = fmaxf(acc[i][j][r] + bias, 0.f);
      }
  }
}

// ---------------------------------------------------------------------------
// K8: in-place log_softmax over V per row; row cached in registers (1 read)
// grid B, block 1024
// ---------------------------------------------------------------------------
__global__ void k_logsoftmax(float* __restrict__ out_logits) {
  const int b = blockIdx.x;
  const int t = threadIdx.x;  // 1024
  float* row = out_logits + (size_t)b * V_;
  __shared__ float red[1024];

  float lv[32];
  float m = -INFINITY;
#pragma unroll
  for (int j = 0; j < 32; j++) {
    const int i = j * 1024 + t;
    lv[j] = (i < V_) ? row[i] : -INFINITY;
    m = fmaxf(m, lv[j]);
  }
  red[t] = m; __syncthreads();
  for (int off = 512; off; off >>= 1) {
    if (t < off) red[t] = fmaxf(red[t], red[t + off]);
    __syncthreads();
  }
  const float lm = red[0]; __syncthreads();

  float s = 0.f;
#pragma unroll
  for (int j = 0; j < 32; j++) {
    const int i = j * 1024 + t;
    if (i < V_) s += __expf(lv[j] - lm);
  }
  red[t] = s; __syncthreads();
  for (int off = 512; off; off >>= 1) {
    if (t < off) red[t] += red[t + off];
    __syncthreads();
  }
  const float lse = lm + __logf(red[0]);

#pragma unroll
  for (int j = 0; j < 32; j++) {
    const int i = j * 1024 + t;
    if (i < V_) row[i] = lv[j] - lse;
  }
}

// ---------------------------------------------------------------------------
extern "C" void kernel_launch(void* const* d_in, const int* in_sizes, int n_in,
                              void* d_out, int out_size, void* d_ws, size_t ws_size,
                              hipStream_t stream) {
  (void)in_sizes; (void)n_in; (void)out_size; (void)ws_size;

  const int*   tokens   = (const int*)  d_in[0];
  const float* enc      = (const float*)d_in[1];
  const float* hidden   = (const float*)d_in[2];
  const float* cell     = (const float*)d_in[3];
  const float* emb      = (const float*)d_in[4];
  const float* W_energy = (const float*)d_in[5];
  const float* b_energy = (const float*)d_in[6];
  const float* w_value  = (const float*)d_in[7];
  const float* W_ih     = (const float*)d_in[8];
  const float* W_hh     = (const float*)d_in[9];
  const float* b_ih     = (const float*)d_in[10];
  const float* b_hh     = (const float*)d_in[11];
  const float* W_out    = (const float*)d_in[12];
  const float* b_out    = (const float*)d_in[13];

  // outputs concatenated flat: log_softmax [B,1,V], h_new [1,B,H],
  // c_new [1,B,H], attn [B,1,S]
  float* out        = (float*)d_out;
  float* out_logits = out;
  float* out_h      = out + (size_t)B_ * V_;
  float* out_c      = out_h + B_ * H_;
  float* out_attn   = out_c + B_ * H_;

  // workspace: 4 MiB of f32
  float* ws       = (float*)d_ws;
  float* ws_u     = ws;                    // B*H    = 131072
  float* ws_score = ws_u + B_ * H_;        // B*S    =  65536
  float* ws_x     = ws_score + B_ * S_;    // B*1280 = 327680
  float* ws_gates = ws_x + B_ * 1280;      // B*2048 = 524288

  k_embed_prep <<<B_, 256, 0, stream>>>(tokens, hidden, emb, ws_x);
  k_gemm_u     <<<dim3(2, 8), 128, 0, stream>>>(hidden, W_energy, b_energy, ws_u);
  k_score      <<<(B_ * S_) / 16, 256, 0, stream>>>(enc, W_energy, ws_u, w_value, ws_score);
  k_softmax_ctx<<<B_, 256, 0, stream>>>(ws_score, enc, out_attn, ws_x);
  k_gemm_gates <<<dim3(8, 8), 128, 0, stream>>>(ws_x, W_ih, W_hh, b_ih, b_hh, ws_gates);
  k_lstm       <<<(B_ * H_) / 256, 256, 0, stream>>>(ws_gates, cell, out_h, out_c);
  k_gemm_logits<<<dim3(125, 8), 128, 0, stream>>>(out_h, W_out, b_out, out_logits);
  k_logsoftmax <<<B_, 1024, 0, stream>>>(out_logits);
}